// BPMLLLoss_86998857548253
// MI455X (gfx1250) — compile-verified
//
#include <hip/hip_runtime.h>
#include <math.h>

typedef __attribute__((ext_vector_type(2))) float v2f;
typedef __attribute__((ext_vector_type(8))) float v8f;

// Exact fp32 wave(32)-wide sum using V_WMMA_F32_16X16X4_F32.
// A layout (16x4, 32-bit): lanes 0-15 hold (M=lane, K=0..1) in VGPR0/1,
// lanes 16-31 hold (M=lane-16, K=2..3). We place the per-lane partial in the
// first K slot and zero the second; B = ones(4x16). Then
//   D[m][n] = partial(lane m) + partial(lane m+16)   for every n.
// D layout: VGPR r on lanes 0-15 = D[r][lane] = s_r, lanes 16-31 = s_{r+8}.
// Summing the 8 accumulator regs gives a half sum; one xor-16 permute adds
// the other half. Result (full wave sum) is uniform across the wave.
__device__ __forceinline__ float wave_sum_wmma(float x) {
  v2f a;
  a.x = x;
  a.y = 0.0f;
  v2f b;
  b.x = 1.0f;
  b.y = 1.0f;
  v8f c = {};
  c = __builtin_amdgcn_wmma_f32_16x16x4_f32(
      /*neg_a=*/false, a, /*neg_b=*/false, b,
      /*c_mod=*/(short)0, c, /*reuse_a=*/false, /*reuse_b=*/false);
  float t = c[0] + c[1] + c[2] + c[3] + c[4] + c[5] + c[6] + c[7];
  t += __shfl_xor(t, 16, 32);
  return t;
}

__global__ void __launch_bounds__(256) bpmll_zero_kernel(float* out) {
  out[0] = 0.0f;
}

// One block per batch row. Each thread strides over the row (coalesced),
// accumulating:
//   p   = sum over positive labels of exp(-x)
//   n   = sum over negative labels of exp(+x)
//   cnt = number of positive labels
// Row loss = p*n / (cnt*(L-cnt)); atomically added into the scalar output.
__global__ void __launch_bounds__(256) bpmll_loss_kernel(
    const float* __restrict__ input, const int* __restrict__ target,
    float* __restrict__ out, int L) {
  const int b = blockIdx.x;
  const int tid = threadIdx.x;
  const float* __restrict__ row = input + (size_t)b * (size_t)L;
  const int* __restrict__ trow = target + (size_t)b * (size_t)L;

  float p = 0.0f;    // sum exp(-x) over positives
  float n = 0.0f;    // sum exp(+x) over negatives
  float cnt = 0.0f;  // number of positives (exact in fp32, max 1024)

  for (int i = tid; i < L; i += blockDim.x) {
    const float x = row[i];
    const int y = trow[i];
    if (y == 1) {
      p += expf(-x);
      cnt += 1.0f;
    } else {
      n += expf(x);
    }
  }

  // Uniform control flow from here: every thread in the block participates,
  // so EXEC is all-ones at the WMMA call sites.
  p = wave_sum_wmma(p);
  n = wave_sum_wmma(n);
  cnt = wave_sum_wmma(cnt);

  __shared__ float sp[8], sn[8], sc[8];
  const int wid = tid >> 5;
  const int lane = tid & 31;
  if (lane == 0) {
    sp[wid] = p;
    sn[wid] = n;
    sc[wid] = cnt;
  }
  __syncthreads();

  if (tid == 0) {
    float P = 0.0f, N = 0.0f, C = 0.0f;
    const int nw = (int)(blockDim.x >> 5);
    for (int w = 0; w < nw; ++w) {
      P += sp[w];
      N += sn[w];
      C += sc[w];
    }
    const float len = C * ((float)L - C);  // npos * nneg
    const float loss = (P * N) / len;      // 0/0 -> NaN, matching reference
    atomicAdd(out, loss);
  }
}

extern "C" void kernel_launch(void* const* d_in, const int* in_sizes, int n_in,
                              void* d_out, int out_size, void* d_ws,
                              size_t ws_size, hipStream_t stream) {
  (void)n_in;
  (void)out_size;
  (void)d_ws;
  (void)ws_size;

  const float* input = (const float*)d_in[0];
  const int* target = (const int*)d_in[1];
  float* out = (float*)d_out;

  const int L = 1024;
  const int B = in_sizes[0] / L;

  bpmll_zero_kernel<<<1, 1, 0, stream>>>(out);
  bpmll_loss_kernel<<<B, 256, 0, stream>>>(input, target, out, L);
}